// CNN_20435454394649
// MI455X (gfx1250) — compile-verified
//
#include <hip/hip_runtime.h>

typedef __attribute__((ext_vector_type(2))) float v2f;
typedef __attribute__((ext_vector_type(8))) float v8f;

#define L_SIG   900000      // 3000 tokens * 300 dims, flattened signal length
#define CHUNK   4096        // signal positions per block
#define LDSN    (CHUNK + 16)
#define NTHREADS 256        // 8 wave32 waves
#define NWAVES  8
#define TILES   (CHUNK / 16)
#define NGRP    21          // 3 convs * ceil(100/16)=7 groups of 16 filters
#define NTOK    16          // token slots covered by one chunk (<=15)

// ---------------------------------------------------------------------------
// Tile sweep: A fragment from LDS once per 16-position tile, then one (or two,
// conv3) V_WMMA_F32_16X16X4_F32 per 16-filter group; fold the 8 accumulator
// VGPRs into a per-group running max. MASKED=true only for the final position
// block (tail validity: t <= L-K; invalid -> 0, harmless since max(ReLU) >= 0).
// ---------------------------------------------------------------------------
template<bool MASKED>
__device__ __forceinline__ void conv_sweep(const float* __restrict__ lds_h,
                                           int base, int wave, int nlo, int hi,
                                           int koff,
                                           const v2f (&bf)[NGRP],
                                           const v2f (&b2f)[7],
                                           const float (&biasl)[NGRP],
                                           float (&mymax)[NGRP])
{
    for (int tile = wave; tile < TILES; tile += NWAVES) {
        int t16 = tile * 16;
        int ai  = t16 + nlo + koff;     // A[m=nlo, k=koff..koff+1] = h[t0+m+k]
        v2f a, a2;
        a.x  = lds_h[ai + 0];
        a.y  = lds_h[ai + 1];
        a2.x = lds_h[ai + 4];           // taps 4..7 slab (conv3 only)
        a2.y = lds_h[ai + 5];

        #pragma unroll
        for (int g = 0; g < NGRP; ++g) {
            float bl = biasl[g];
            v8f c = { bl, bl, bl, bl, bl, bl, bl, bl };   // bias folded into C
            c = __builtin_amdgcn_wmma_f32_16x16x4_f32(
                    false, a, false, bf[g], (short)0, c, false, false);
            if (g >= 14)
                c = __builtin_amdgcn_wmma_f32_16x16x4_f32(
                        false, a2, false, b2f[g - 14], (short)0, c, false, false);

            // C layout: VGPR r, lane -> M = r + 8*hi, N = nlo.
            if (MASKED) {
                int K  = 3 + g / 7;
                int tl = L_SIG - K;                // last valid output position
                int p0 = base + t16 + hi * 8;
                #pragma unroll
                for (int r = 0; r < 8; ++r) {
                    float y = ((p0 + r) <= tl) ? c[r] : 0.0f;
                    mymax[g] = fmaxf(mymax[g], y);
                }
            } else {
                #pragma unroll
                for (int r = 0; r < 8; ++r)
                    mymax[g] = fmaxf(mymax[g], c[r]);
            }
        }
    }
}

// ---------------------------------------------------------------------------
// Fused: token prefetch -> async embedding gather (global -> LDS, ASYNCcnt)
//        -> conv(K=3,4,5) via V_WMMA_F32_16X16X4_F32 -> bias+ReLU+max
//        -> global atomicMax commit of per-filter features.
// ---------------------------------------------------------------------------
__global__ __launch_bounds__(NTHREADS)
void textcnn_conv_wmma(const int* __restrict__ x,
                       const float* __restrict__ emb,
                       const float* __restrict__ w1, const float* __restrict__ b1,
                       const float* __restrict__ w2, const float* __restrict__ b2,
                       const float* __restrict__ w3, const float* __restrict__ b3,
                       unsigned int* __restrict__ featbits)
{
    __shared__ float lds_h[LDSN];
    __shared__ int   tok_lds[NTOK];
    const int b    = blockIdx.y;
    const int base = blockIdx.x * CHUNK;
    const int tid  = threadIdx.x;
    const int s0   = base / 300;        // first token slot this chunk touches

    // ---- stage the <=15 token ids this chunk needs ----
    if (tid < NTOK) {
        int s = s0 + tid;
        tok_lds[tid] = (s < 3000) ? x[b * 3000 + s] : 0;
    }
    __syncthreads();

    // ---- async gather embedding words straight into LDS (zero-pad tail) ----
    {
        unsigned lbase = (unsigned)(uintptr_t)lds_h;   // LDS byte offset base
        for (int i = tid; i < LDSN; i += NTHREADS) {
            int p = base + i;
            if (p < L_SIG) {
                int s   = p / 300;
                int d   = p - s * 300;
                int tok = tok_lds[s - s0];
                unsigned long long ga =
                    (unsigned long long)(uintptr_t)(emb + (long)tok * 300 + d);
                unsigned la = lbase + (unsigned)i * 4u;
                asm volatile("global_load_async_to_lds_b32 %0, %1, off"
                             :: "v"(la), "v"(ga) : "memory");
            } else {
                lds_h[i] = 0.0f;        // finite zeros for padded taps
            }
        }
        asm volatile("s_wait_asynccnt 0x0" ::: "memory");
    }
    __syncthreads();

    const int lane = tid & 31;
    const int wave = tid >> 5;
    const int nlo  = lane & 15;   // filter column / A-matrix row M
    const int hi   = lane >> 4;   // 0: lanes 0-15, 1: lanes 16-31
    const int koff = hi * 2;      // K rows handled by this half-wave

    // ---- per-lane weight (B) fragments, bias, running maxima in registers ----
    v2f   bf[NGRP];     // B rows {koff, koff+1} for taps 0..3
    v2f   b2f[7];       // conv3 second WMMA: only tap k=4 live (rows 4..7)
    float biasl[NGRP];
    float mymax[NGRP];

    const float* wptr[3] = { w1, w2, w3 };
    const float* bptr[3] = { b1, b2, b3 };

    #pragma unroll
    for (int g = 0; g < NGRP; ++g) {
        int conv = g / 7;
        int K    = 3 + conv;
        int f    = (g % 7) * 16 + nlo;
        bool fv  = (f < 100);
        const float* wc = wptr[conv];
        bf[g].x  = (fv && (koff + 0) < K) ? wc[f * K + koff + 0] : 0.0f;
        bf[g].y  = (fv && (koff + 1) < K) ? wc[f * K + koff + 1] : 0.0f;
        biasl[g] = fv ? bptr[conv][f] : 0.0f;
        mymax[g] = 0.0f;                        // ReLU floor
        if (conv == 2) {
            b2f[g - 14].x = (fv && hi == 0) ? wc[f * 5 + 4] : 0.0f; // row k=4
            b2f[g - 14].y = 0.0f;                                    // rows 5..7
        }
    }

    // ---- sweep tiles: only the final block needs tail-position masking ----
    if (base + CHUNK + 4 <= L_SIG)
        conv_sweep<false>(lds_h, base, wave, nlo, hi, koff, bf, b2f, biasl, mymax);
    else
        conv_sweep<true >(lds_h, base, wave, nlo, hi, koff, bf, b2f, biasl, mymax);

    // ---- reduce M=0..7 vs M=8..15 halves, commit via monotone uint atomicMax ----
    #pragma unroll
    for (int g = 0; g < NGRP; ++g) {
        float v = mymax[g];
        v = fmaxf(v, __shfl_xor(v, 16, 32));   // wave32 cross-half max
        int conv = g / 7;
        int f    = (g % 7) * 16 + nlo;
        if (lane < 16 && f < 100) {
            // v >= 0 always -> IEEE bit pattern is order-preserving
            atomicMax(&featbits[b * 300 + conv * 100 + f], __float_as_uint(v));
        }
    }
}

__global__ void zero_feats(unsigned int* __restrict__ p, int n)
{
    int i = blockIdx.x * blockDim.x + threadIdx.x;
    if (i < n) p[i] = 0u;
}

// feats[2,300] @ fc_w[10,300]^T + fc_b -> out[2,10]
__global__ void textcnn_fc(const unsigned int* __restrict__ featbits,
                           const float* __restrict__ fc_w,
                           const float* __restrict__ fc_b,
                           float* __restrict__ out)
{
    int t = threadIdx.x;
    if (t < 20) {
        int b = t / 10, j = t % 10;
        float s = fc_b[j];
        for (int i = 0; i < 300; ++i)
            s += __uint_as_float(featbits[b * 300 + i]) * fc_w[j * 300 + i];
        out[b * 10 + j] = s;
    }
}

extern "C" void kernel_launch(void* const* d_in, const int* in_sizes, int n_in,
                              void* d_out, int out_size, void* d_ws, size_t ws_size,
                              hipStream_t stream)
{
    (void)in_sizes; (void)n_in; (void)out_size; (void)ws_size;
    const int*   x    = (const int*)  d_in[0];
    const float* emb  = (const float*)d_in[1];
    const float* w1   = (const float*)d_in[2];
    const float* b1   = (const float*)d_in[3];
    const float* w2   = (const float*)d_in[4];
    const float* b2   = (const float*)d_in[5];
    const float* w3   = (const float*)d_in[6];
    const float* b3   = (const float*)d_in[7];
    const float* fc_w = (const float*)d_in[8];
    const float* fc_b = (const float*)d_in[9];
    unsigned int* featbits = (unsigned int*)d_ws;

    zero_feats<<<dim3(3), dim3(256), 0, stream>>>(featbits, 2 * 300);

    dim3 grid((L_SIG + CHUNK - 1) / CHUNK, 2);
    textcnn_conv_wmma<<<grid, dim3(NTHREADS), 0, stream>>>(
        x, emb, w1, b1, w2, b2, w3, b3, featbits);

    textcnn_fc<<<dim3(1), dim3(32), 0, stream>>>(featbits, fc_w, fc_b, (float*)d_out);
}